// SoftAggBasic_8873402434227
// MI455X (gfx1250) — compile-verified
//
#include <hip/hip_runtime.h>
#include <math.h>

typedef __attribute__((ext_vector_type(2))) float v2f;
typedef __attribute__((ext_vector_type(8))) float v8f;

#define GMAX 512   // NUM_GROUPS upper bound from the reference

// ---------------------------------------------------------------- bucketing
__global__ void init_counts(int* count, int* cursor) {
  int t = blockIdx.x * blockDim.x + threadIdx.x;
  if (t < GMAX) { count[t] = 0; cursor[t] = 0; }
}

__global__ void count_groups(const int* __restrict__ ix, int* __restrict__ count, int N) {
  int n = blockIdx.x * blockDim.x + threadIdx.x;
  if (n < N) atomicAdd(&count[ix[n]], 1);
}

// single block, GMAX threads: exclusive prefix sum (Hillis-Steele in LDS)
__global__ void scan_offsets(const int* __restrict__ count, int* __restrict__ offs) {
  __shared__ int tmp[GMAX];
  int t = threadIdx.x;
  tmp[t] = count[t];
  __syncthreads();
  for (int off = 1; off < GMAX; off <<= 1) {
    int add = (t >= off) ? tmp[t - off] : 0;
    __syncthreads();
    tmp[t] += add;
    __syncthreads();
  }
  offs[t] = tmp[t] - count[t];
  if (t == GMAX - 1) offs[GMAX] = tmp[t];
}

__global__ void scatter_buckets(const int* __restrict__ ix, const int* __restrict__ offs,
                                int* __restrict__ cursor, int* __restrict__ bucket, int N) {
  int n = blockIdx.x * blockDim.x + threadIdx.x;
  if (n < N) {
    int g = ix[n];
    int p = offs[g] + atomicAdd(&cursor[g], 1);
    bucket[p] = n;
  }
}

// ---------------------------------------------------------------- scores: s[b,n] = x[b,n,:].Wg + bg
// one wave (32 lanes) per (b,n) row; float4 loads -> 4 b128 VMEM ops per row
__global__ void score_kernel(const float* __restrict__ x, const float* __restrict__ Wg,
                             const float* __restrict__ bg, float* __restrict__ s,
                             int Ntot, int D) {
  int wave = (blockIdx.x * blockDim.x + threadIdx.x) >> 5;
  int lane = threadIdx.x & 31;
  if (wave >= Ntot) return;
  const float4* row = (const float4*)(x + (size_t)wave * D);
  const float4* wg4 = (const float4*)Wg;
  int D4 = D >> 2;
  float acc = 0.f;
  for (int i = lane; i < D4; i += 32) {
    float4 xv = row[i];
    float4 wv = wg4[i];
    acc += xv.x * wv.x + xv.y * wv.y + xv.z * wv.z + xv.w * wv.w;
  }
  for (int off = 16; off > 0; off >>= 1) acc += __shfl_xor(acc, off, 32);
  if (lane == 0) s[wave] = acc + bg[0];
}

// ---------------------------------------------------------------- per-group softmax stats (max, sum-exp)
__global__ void stats_kernel(const float* __restrict__ s, const int* __restrict__ bucket,
                             const int* __restrict__ offs, float* __restrict__ m,
                             float* __restrict__ denom, int N, int B) {
  int g = blockIdx.x, b = blockIdx.y;
  int beg = offs[g], end = offs[g + 1];
  __shared__ float red[128];
  int t = threadIdx.x;
  float mx = -INFINITY;
  for (int i = beg + t; i < end; i += 128) mx = fmaxf(mx, s[(size_t)b * N + bucket[i]]);
  red[t] = mx; __syncthreads();
  for (int off = 64; off > 0; off >>= 1) {
    if (t < off) red[t] = fmaxf(red[t], red[t + off]);
    __syncthreads();
  }
  mx = red[0]; __syncthreads();
  float sum = 0.f;
  for (int i = beg + t; i < end; i += 128) sum += expf(s[(size_t)b * N + bucket[i]] - mx);
  red[t] = sum; __syncthreads();
  for (int off = 64; off > 0; off >>= 1) {
    if (t < off) red[t] += red[t + off];
    __syncthreads();
  }
  if (t == 0) { m[g * B + b] = mx; denom[g * B + b] = red[0]; }
}

// ---------------------------------------------------------------- xbar[b,g,:] = sum_n_in_g w_n * x[b,n,:]
// one 128-thread block per (g,b); D=512 -> each thread owns one float4 (b128 coalesced rows)
__global__ void xbar_kernel(const float* __restrict__ x, const float* __restrict__ s,
                            const int* __restrict__ bucket, const int* __restrict__ offs,
                            const float* __restrict__ m, const float* __restrict__ denom,
                            float* __restrict__ xbar, int N, int B, int G) {
  const int D = 512;
  int g = blockIdx.x, b = blockIdx.y;
  int beg = offs[g], end = offs[g + 1];
  int t = threadIdx.x;                    // 0..127, owns d = [4t, 4t+4)
  float mg = m[g * B + b];
  float inv = (end > beg) ? 1.0f / denom[g * B + b] : 0.f;
  float4 acc = make_float4(0.f, 0.f, 0.f, 0.f);
  for (int i = beg; i < end; ++i) {
    int n = bucket[i];
    float w = expf(s[(size_t)b * N + n] - mg) * inv;          // broadcast load + v_exp
    const float4* row = (const float4*)(x + ((size_t)b * N + n) * D);
    float4 xv = row[t];                                        // coalesced 2KB row
    acc.x += w * xv.x; acc.y += w * xv.y;
    acc.z += w * xv.z; acc.w += w * xv.w;
  }
  float4* o = (float4*)(xbar + ((size_t)b * G + g) * D);
  o[t] = acc;
}

// ---------------------------------------------------------------- f32 WMMA GEMM: C[M,N] = A[M,K] * B[K,N] + bias
// One wave computes a 32x32 tile (2x2 of 16x16), V_WMMA_F32_16X16X4_F32, K-step 4.
// Fragment layouts per CDNA5 ISA 7.12.2:
//   A (16x4, 2 VGPR): lane&15 = M row; lane>=16 selects K+2; vgpr = K parity
//   B (4x16, 2 VGPR): lane&15 = N col; lane>=16 selects K+2; vgpr = K parity
//   C/D (16x16, 8 VGPR): vgpr j -> row j (+8 for lanes 16-31), lane&15 = col
__global__ void gemm_wmma_f32(const float* __restrict__ A, const float* __restrict__ Bm,
                              const float* __restrict__ bias, float* __restrict__ C,
                              int M, int N, int K) {
  int lane = threadIdx.x & 31;
  int wave_id = (blockIdx.x * blockDim.x + threadIdx.x) >> 5;
  int tiles_n = N >> 5;
  int total = (M >> 5) * tiles_n;
  if (wave_id >= total) return;            // wave-uniform: EXEC stays all-ones
  int row0 = (wave_id / tiles_n) << 5;
  int col0 = (wave_id % tiles_n) << 5;

  int mrow = lane & 15;
  int koff = (lane >> 4) << 1;             // 0 or 2

  v8f c00 = {}, c01 = {}, c10 = {}, c11 = {};
  for (int k = 0; k < K; k += 4) {
    const float* ap0 = A + (size_t)(row0 + mrow) * K + (k + koff);
    const float* ap1 = ap0 + (size_t)16 * K;
    v2f a0; a0.x = ap0[0]; a0.y = ap0[1];
    v2f a1; a1.x = ap1[0]; a1.y = ap1[1];
    const float* bp = Bm + (size_t)(k + koff) * N + col0 + mrow;
    v2f b0; b0.x = bp[0];  b0.y = bp[N];
    v2f b1; b1.x = bp[16]; b1.y = bp[N + 16];
    c00 = __builtin_amdgcn_wmma_f32_16x16x4_f32(false, a0, false, b0, (short)0, c00, false, false);
    c01 = __builtin_amdgcn_wmma_f32_16x16x4_f32(false, a0, false, b1, (short)0, c01, false, false);
    c10 = __builtin_amdgcn_wmma_f32_16x16x4_f32(false, a1, false, b0, (short)0, c10, false, false);
    c11 = __builtin_amdgcn_wmma_f32_16x16x4_f32(false, a1, false, b1, (short)0, c11, false, false);
  }

  int rsel = (lane >> 4) << 3;             // 0 or 8
  int cc = col0 + mrow;
  float bi0 = bias ? bias[cc] : 0.f;
  float bi1 = bias ? bias[cc + 16] : 0.f;
#pragma unroll
  for (int j = 0; j < 8; ++j) {
    int r = row0 + j + rsel;
    C[(size_t)r * N + cc]             = c00[j] + bi0;
    C[(size_t)r * N + cc + 16]        = c01[j] + bi1;
    C[(size_t)(r + 16) * N + cc]      = c10[j] + bi0;
    C[(size_t)(r + 16) * N + cc + 16] = c11[j] + bi1;
  }
}

// ---------------------------------------------------------------- out[b,n,:] = hy[b, ix[n], :]
__global__ void gather_kernel(const float* __restrict__ hy, const int* __restrict__ ix,
                              float* __restrict__ out, int N, int B, int G, int D) {
  int D4 = D >> 2;
  size_t idx = (size_t)blockIdx.x * blockDim.x + threadIdx.x;
  size_t total = (size_t)B * N * D4;
  if (idx >= total) return;
  int d4 = (int)(idx % D4);
  size_t bn = idx / D4;
  int n = (int)(bn % N);
  int b = (int)(bn / N);
  const float4* src = (const float4*)(hy + ((size_t)b * G + ix[n]) * D);
  float4* dst = (float4*)(out + bn * D);
  dst[d4] = src[d4];
}

// ---------------------------------------------------------------- host
extern "C" void kernel_launch(void* const* d_in, const int* in_sizes, int n_in,
                              void* d_out, int out_size, void* d_ws, size_t ws_size,
                              hipStream_t stream) {
  const float* x  = (const float*)d_in[0];
  const int*   ix = (const int*)d_in[1];   // jnp "int64" is int32 under default JAX config
  const float* Wf = (const float*)d_in[2];
  const float* bf = (const float*)d_in[3];
  const float* Wg = (const float*)d_in[4];
  const float* bg = (const float*)d_in[5];
  const float* Wh = (const float*)d_in[6];
  const float* bh = (const float*)d_in[7];
  float* out = (float*)d_out;

  const int N = in_sizes[1];
  const int D = 512;
  const int B = in_sizes[0] / (N * D);
  const int G = GMAX;

  char* ws = (char*)d_ws;
  size_t off = 0;
  auto carve = [&](size_t bytes) -> void* {
    void* p = ws + off;
    off = (off + bytes + 255) & ~(size_t)255;
    return p;
  };
  float* s      = (float*)carve((size_t)B * N * 4);
  int*   count  = (int*)  carve((size_t)G * 4);
  int*   offs   = (int*)  carve((size_t)(G + 1) * 4);
  int*   cursor = (int*)  carve((size_t)G * 4);
  int*   bucket = (int*)  carve((size_t)N * 4);
  float* m      = (float*)carve((size_t)G * B * 4);
  float* denom  = (float*)carve((size_t)G * B * 4);
  float* xbar   = (float*)carve((size_t)B * G * D * 4);
  float* y      = (float*)carve((size_t)B * G * D * 4);
  float* hy     = (float*)carve((size_t)B * G * D * 4);
  (void)ws_size;

  // 1) buckets (counting sort by group)
  init_counts<<<(G + 255) / 256, 256, 0, stream>>>(count, cursor);
  count_groups<<<(N + 255) / 256, 256, 0, stream>>>(ix, count, N);
  scan_offsets<<<1, GMAX, 0, stream>>>(count, offs);
  scatter_buckets<<<(N + 255) / 256, 256, 0, stream>>>(ix, offs, cursor, bucket, N);

  // 2) scores (x pass #1, float4 loads)
  {
    int Ntot = B * N;                       // one wave per row
    long long threads = (long long)Ntot * 32;
    score_kernel<<<(unsigned)((threads + 255) / 256), 256, 0, stream>>>(x, Wg, bg, s, Ntot, D);
  }

  // 3) segment-softmax stats
  stats_kernel<<<dim3(G, B), 128, 0, stream>>>(s, bucket, offs, m, denom, N, B);

  // 4) weighted group means xbar (x pass #2, float4) — no float atomics
  xbar_kernel<<<dim3(G, B), 128, 0, stream>>>(x, s, bucket, offs, m, denom, xbar, N, B, G);

  // 5) y = xbar*Wf + bf ; hy = y*Wh + bh  (f32 WMMA GEMMs)
  {
    int M1 = B * G;
    int waves = (M1 >> 5) * (D >> 5);
    gemm_wmma_f32<<<(waves + 3) / 4, 128, 0, stream>>>(xbar, Wf, bf, y, M1, D, D);
    gemm_wmma_f32<<<(waves + 3) / 4, 128, 0, stream>>>(y, Wh, bh, hy, M1, D, D);
  }

  // 6) gather back to [B,N,D]
  {
    size_t total = (size_t)B * N * (D / 4);
    gather_kernel<<<(unsigned)((total + 255) / 256), 256, 0, stream>>>(hy, ix, out, N, B, G, D);
  }
}